// oriRGCN_12738873000206
// MI455X (gfx1250) — compile-verified
//
#include <hip/hip_runtime.h>

// ---------------------------------------------------------------------------
// RGCN fused pipeline for MI455X (gfx1250, wave32, WMMA)
//   k0: zero per-relation accumulators
//   k1: pre-swizzle block-diagonal [W_des|W_tweet] into f32-WMMA B fragments
//   k2: fused feature MLPs via V_WMMA_F32_16X16X4_F32 -> x[N,32], root -> out
//   k3: edge gather + per-relation message + atomic mean accumulation
//   k4: finalize out += sum_r s_r / max(cnt_r, 1)
// ---------------------------------------------------------------------------

typedef float v2f __attribute__((ext_vector_type(2)));
typedef float v8f __attribute__((ext_vector_type(8)));

#define KSTEPS 384          // 1536 / 4 (des 768 + tweet 768, K-step = 4)
#define ROWS_PER_BLOCK 128  // 8 waves x 16 rows

__global__ __launch_bounds__(256) void k_zero(float* __restrict__ p, int n) {
  int i = blockIdx.x * 256 + threadIdx.x;
  if (i < n) p[i] = 0.0f;
}

// Build per-K-step WMMA B fragments. Fragment for step s, lane l:
//   col n = l&15, kbase = (l>>4)*2, k = s*4 + kbase
//   value pair = Bpad[k][n], Bpad[k+1][n]
// Bpad[0:768, 0:8] = W_des, Bpad[768:1536, 8:16] = W_tweet, zero elsewhere.
__global__ __launch_bounds__(256) void k_bfrag(const float* __restrict__ Wd,
                                               const float* __restrict__ Wt,
                                               float* __restrict__ bf) {
  int idx = blockIdx.x * 256 + threadIdx.x;   // over KSTEPS*32 (s, lane)
  if (idx >= KSTEPS * 32) return;
  int s = idx >> 5;
  int l = idx & 31;
  int n = l & 15;
  int k = s * 4 + ((l >> 4) << 1);
  float v0 = 0.0f, v1 = 0.0f;
  if (k < 768) {
    if (n < 8) { v0 = Wd[k * 8 + n]; v1 = Wd[(k + 1) * 8 + n]; }
  } else {
    int kk = k - 768;
    if (n >= 8) { v0 = Wt[kk * 8 + (n - 8)]; v1 = Wt[(kk + 1) * 8 + (n - 8)]; }
  }
  bf[idx * 2 + 0] = v0;
  bf[idx * 2 + 1] = v1;
}

// Fused MLP: each wave computes a 16-row tile of x[N,32] with f32 WMMA over
// K=1536 (des then tweet), plus num/cat columns via VALU. Tile is transposed
// through LDS for fully coalesced x-row stores; root transform written to out.
__global__ __launch_bounds__(256) void k_mlp(
    const float* __restrict__ des, const float* __restrict__ tweet,
    const float* __restrict__ nump, const float* __restrict__ catp,
    const float* __restrict__ b_des, const float* __restrict__ b_tweet,
    const float* __restrict__ W_num, const float* __restrict__ b_num,
    const float* __restrict__ W_cat, const float* __restrict__ b_cat,
    const float* __restrict__ W_root, const float* __restrict__ bfrag,
    float* __restrict__ x, float* __restrict__ out, int N) {
  __shared__ float xtile[ROWS_PER_BLOCK * 32];   // 16 KB

  const int lane = threadIdx.x & 31;
  const int wave = threadIdx.x >> 5;
  const int n    = lane & 15;        // A row index == C column index
  const int hi   = lane >> 4;        // K-pair / C row-half selector

  int node = (blockIdx.x * 8 + wave) * 16;
  if (node > N - 16) node = N - 16;  // N is a multiple of 16; tail waves redo

  const float* aDes = des   + (size_t)(node + n) * 768 + hi * 2;
  const float* aTwt = tweet + (size_t)(node + n) * 768 + hi * 2;
  const v2f*   bf   = (const v2f*)bfrag;

  v8f c = {};
#pragma unroll 4
  for (int s = 0; s < 192; ++s) {
    v2f a = *(const v2f*)(aDes + s * 4);
    v2f b = bf[s * 32 + lane];
    c = __builtin_amdgcn_wmma_f32_16x16x4_f32(false, a, false, b,
                                              (short)0, c, false, false);
  }
#pragma unroll 4
  for (int s = 192; s < 384; ++s) {
    v2f a = *(const v2f*)(aTwt + (s - 192) * 4);
    v2f b = bf[s * 32 + lane];
    c = __builtin_amdgcn_wmma_f32_16x16x4_f32(false, a, false, b,
                                              (short)0, c, false, false);
  }

  // Bias + leaky_relu on the 16 WMMA columns; C row for vgpr j is j + 8*hi.
  const float bias = (n < 8) ? b_des[n] : b_tweet[n - 8];
#pragma unroll
  for (int j = 0; j < 8; ++j) {
    float v = c[j] + bias;
    v = (v > 0.0f) ? v : 0.01f * v;
    xtile[(wave * 16 + j + 8 * hi) * 32 + n] = v;
  }

  // num/cat columns (16..31): lane handles column 16+n for its 8 rows.
  {
    const int q = (n < 8) ? n : (n - 8);
#pragma unroll
    for (int j = 0; j < 8; ++j) {
      const int row = node + j + 8 * hi;
      float v;
      if (n < 8) {
        const float* np = nump + (size_t)row * 5;
        v = b_num[q];
#pragma unroll
        for (int t = 0; t < 5; ++t) v += np[t] * W_num[t * 8 + q];
      } else {
        v = catp[row] * W_cat[q] + b_cat[q];
      }
      v = (v > 0.0f) ? v : 0.01f * v;
      xtile[(wave * 16 + j + 8 * hi) * 32 + 16 + n] = v;
    }
  }

  __syncthreads();

  // Coalesced x store: each x row is exactly one 128B line (8 x float4).
  const int base = blockIdx.x * ROWS_PER_BLOCK;
  float4* xg = (float4*)x;
  const float4* xt4 = (const float4*)xtile;
  for (int i = threadIdx.x; i < ROWS_PER_BLOCK * 8; i += 256) {
    int row = base + (i >> 3);
    if (row < N) xg[(size_t)row * 8 + (i & 7)] = xt4[i];
  }

  // Root transform: out[row] = x[row] @ W_root, from the LDS tile.
  for (int i = threadIdx.x; i < ROWS_PER_BLOCK * 2; i += 256) {
    int row = base + (i >> 1);
    int o = i & 1;
    if (row < N) {
      const float* xr = &xtile[(i >> 1) * 32];
      float acc = 0.0f;
#pragma unroll
      for (int k = 0; k < 32; ++k) acc += xr[k] * W_root[k * 2 + o];
      out[(size_t)row * 2 + o] = acc;
    }
  }
}

// Edge phase: gather x[src] (L2 resident), per-relation message, atomic sums.
__global__ __launch_bounds__(256) void k_edge(const int* __restrict__ ei,
                                              const int* __restrict__ et,
                                              const float* __restrict__ x,
                                              const float* __restrict__ W_rel,
                                              float* __restrict__ sums,
                                              float* __restrict__ cnts,
                                              int E, int N) {
  __shared__ float wrel[128];   // [2][32][2]
  if (threadIdx.x < 128) wrel[threadIdx.x] = W_rel[threadIdx.x];
  __syncthreads();

  int e = blockIdx.x * 256 + threadIdx.x;
  if (e >= E) return;
  const int s = ei[e];
  const int d = ei[E + e];
  const int r = et[e];

  const float4* xr = (const float4*)(x + (size_t)s * 32);
  const float* w = wrel + r * 64;
  float m0 = 0.0f, m1 = 0.0f;
#pragma unroll
  for (int i = 0; i < 8; ++i) {
    float4 v = xr[i];
    const float* wk = w + 8 * i;
    m0 += v.x * wk[0] + v.y * wk[2] + v.z * wk[4] + v.w * wk[6];
    m1 += v.x * wk[1] + v.y * wk[3] + v.z * wk[5] + v.w * wk[7];
  }
  float* sd = sums + ((size_t)r * N + d) * 2;
  atomicAdd(sd + 0, m0);
  atomicAdd(sd + 1, m1);
  atomicAdd(cnts + (size_t)r * N + d, 1.0f);
}

__global__ __launch_bounds__(256) void k_final(float* __restrict__ out,
                                               const float* __restrict__ sums,
                                               const float* __restrict__ cnts,
                                               int N) {
  int t = blockIdx.x * 256 + threadIdx.x;
  if (t >= N * 2) return;
  const int nd = t >> 1;
  const int o = t & 1;
  float acc = out[t];
#pragma unroll
  for (int r = 0; r < 2; ++r) {
    float cr = cnts[(size_t)r * N + nd];
    acc += sums[((size_t)r * N + nd) * 2 + o] / fmaxf(cr, 1.0f);
  }
  out[t] = acc;
}

extern "C" void kernel_launch(void* const* d_in, const int* in_sizes, int n_in,
                              void* d_out, int out_size, void* d_ws, size_t ws_size,
                              hipStream_t stream) {
  const float* des    = (const float*)d_in[0];
  const float* tweet  = (const float*)d_in[1];
  const float* nump   = (const float*)d_in[2];
  const float* catp   = (const float*)d_in[3];
  const int*   ei     = (const int*)d_in[4];
  const int*   et     = (const int*)d_in[5];
  const float* W_des  = (const float*)d_in[6];
  const float* b_des  = (const float*)d_in[7];
  const float* W_twt  = (const float*)d_in[8];
  const float* b_twt  = (const float*)d_in[9];
  const float* W_num  = (const float*)d_in[10];
  const float* b_num  = (const float*)d_in[11];
  const float* W_cat  = (const float*)d_in[12];
  const float* b_cat  = (const float*)d_in[13];
  const float* W_rel  = (const float*)d_in[14];
  const float* W_root = (const float*)d_in[15];

  const int N = in_sizes[0] / 768;   // 100000
  const int E = in_sizes[5];         // 3200000
  float* out = (float*)d_out;

  // Workspace layout (floats): bfrag | x | sums | cnts
  float* ws    = (float*)d_ws;
  float* bfrag = ws;                                   // KSTEPS*64 = 24576
  float* x     = bfrag + KSTEPS * 64;                  // N*32
  float* sums  = x + (size_t)N * 32;                   // 2*N*2
  float* cnts  = sums + (size_t)4 * N;                 // 2*N

  k_zero<<<(6 * N + 255) / 256, 256, 0, stream>>>(sums, 6 * N);
  k_bfrag<<<(KSTEPS * 32 + 255) / 256, 256, 0, stream>>>(W_des, W_twt, bfrag);
  k_mlp<<<(N + ROWS_PER_BLOCK - 1) / ROWS_PER_BLOCK, 256, 0, stream>>>(
      des, tweet, nump, catp, b_des, b_twt, W_num, b_num, W_cat, b_cat,
      W_root, bfrag, x, out, N);
  k_edge<<<(E + 255) / 256, 256, 0, stream>>>(ei, et, x, W_rel, sums, cnts, E, N);
  k_final<<<(2 * N + 255) / 256, 256, 0, stream>>>(out, sums, cnts, N);
}